// KimiMoEGate_74371653698288
// MI455X (gfx1250) — compile-verified
//
#include <hip/hip_runtime.h>

// ---------------------------------------------------------------------------
// KimiMoEGate fused router for MI455X (gfx1250, wave32, WMMA).
//
// logits[16384,256] = X[16384,2048] * W[256,2048]^T in f32 via
// V_WMMA_F32_16X16X4_F32, fused with sigmoid + group-limited top-k gating.
//
// Roofline: X traffic 128 MB -> ~5.5 us HBM floor @ 23.3 TB/s. Weight (2 MB)
// stays L2-resident; 64-token tiles (256 blocks) hold total weight re-read to
// 512 MB of L2 traffic, and each wave reuses its B fragments across 4
// M-subtiles in registers (8 WMMAs per 2 global-B + 4 LDS-A loads).
//
// Double-buffered A staging via GLOBAL_LOAD_ASYNC_TO_LDS_B128 (ASYNCcnt):
// chunk c+1 streams into LDS while WMMAs consume chunk c. Async loads
// complete in order, so s_wait_asynccnt 8 after issuing the next chunk's
// 8 ops guarantees the previous chunk has landed.
// ---------------------------------------------------------------------------

typedef __attribute__((ext_vector_type(2))) float v2f;
typedef __attribute__((ext_vector_type(8))) float v8f;
typedef int v4i __attribute__((vector_size(16)));           // b128 payload
typedef __attribute__((address_space(1))) v4i* gv4i_p;      // global (as1)
typedef __attribute__((address_space(3))) v4i* lv4i_p;      // LDS (as3)

#define T_TOKENS 16384
#define HDIM     2048
#define NEXP     256
#define NGROUP   8
#define GSIZE    32      // experts per group
#define TOPKG    4
#define TOPK     8
#define TILE_M   64      // tokens per block (4 x 16-row WMMA subtiles)
#define MSUB     4       // TILE_M / 16
#define KCHUNK   128     // K staged per LDS pass
#define NCHUNK   (HDIM / KCHUNK)

#if defined(__has_builtin)
#if __has_builtin(__builtin_amdgcn_global_load_async_to_lds_b128)
#define USE_ASYNC_LDS 1
#endif
#endif
#ifndef USE_ASYNC_LDS
#define USE_ASYNC_LDS 0
#endif

#if defined(__has_builtin)
#if __has_builtin(__builtin_amdgcn_s_wait_asynccnt)
#define WAIT_ASYNC(n) __builtin_amdgcn_s_wait_asynccnt(n)
#endif
#endif
#ifndef WAIT_ASYNC
#define WAIT_ASYNC(n) asm volatile("s_wait_asynccnt %0" :: "n"(n) : "memory")
#endif

__device__ __forceinline__ float sigmoidf_(float x) {
    return 1.0f / (1.0f + __expf(-x));
}

__global__ __launch_bounds__(256)
void kimi_moe_gate_kernel(const float* __restrict__ x,      // [T, H]
                          const float* __restrict__ weight, // [E, H]
                          const float* __restrict__ bias,   // [E]
                          float* __restrict__ out_idx,      // [T, 8] (as float)
                          float* __restrict__ out_w)        // [T, 8]
{
    __shared__ float sA[2 * TILE_M * KCHUNK]; // 64 KB: double-buffered X chunks
    __shared__ float sScores[TILE_M * NEXP];  // 64 KB: sigmoid scores
    __shared__ float sBias[NEXP];             // 1 KB

    const int tid  = threadIdx.x;
    const int wave = tid >> 5;          // 8 wave32s
    const int lane = tid & 31;
    const int half = lane >> 4;         // 0: lanes 0-15, 1: lanes 16-31
    const int l16  = lane & 15;
    const int tokBase = blockIdx.x * TILE_M;

    if (tid < NEXP) sBias[tid] = bias[tid];

    // Wave 'wave' owns experts [32w, 32w+32): two 16-wide N-tiles,
    // and ALL four 16-row M-subtiles of the 64-token tile.
    const int nb = wave * GSIZE;
    const float* wrow0 = weight + (size_t)(nb + l16)      * HDIM;
    const float* wrow1 = weight + (size_t)(nb + 16 + l16) * HDIM;
    const float* xbase = x + (size_t)tokBase * HDIM;

    v8f acc[MSUB][2];
    #pragma unroll
    for (int mt = 0; mt < MSUB; ++mt) { acc[mt][0] = (v8f){}; acc[mt][1] = (v8f){}; }

#if USE_ASYNC_LDS
    // --- Async double-buffered pipeline over K chunks. ---
    // Per chunk per thread: 8 async b128 copies, no VGPR round-trip.
    #define STAGE_ASYNC(CHUNK, BUF)                                            \
        do {                                                                   \
            _Pragma("unroll")                                                  \
            for (int j = 0; j < 8; ++j) {                                      \
                int q   = j * 256 + tid;                                       \
                int row = q >> 5;                                              \
                int c4  = q & 31;                                              \
                gv4i_p gp = (gv4i_p)(xbase + (size_t)row * HDIM +              \
                                     (CHUNK) * KCHUNK + c4 * 4);               \
                lv4i_p lp = (lv4i_p)&sA[(BUF) * TILE_M * KCHUNK +              \
                                        row * KCHUNK + c4 * 4];                \
                __builtin_amdgcn_global_load_async_to_lds_b128(gp, lp, 0, 0);  \
            }                                                                  \
        } while (0)

    STAGE_ASYNC(0, 0);
    for (int c = 0; c < NCHUNK; ++c) {
        const int cur = c & 1;
        const int kc  = c * KCHUNK;
        if (kc + KCHUNK < HDIM) {
            __builtin_prefetch(wrow0 + kc + KCHUNK, 0, 3);
            __builtin_prefetch(wrow1 + kc + KCHUNK, 0, 3);
        }
        if (c + 1 < NCHUNK) {
            STAGE_ASYNC(c + 1, cur ^ 1);
            WAIT_ASYNC(8);   // in-order completion -> chunk c has landed
        } else {
            WAIT_ASYNC(0);
        }
        __syncthreads();     // publish chunk c's LDS data to all waves
        const float* Ab = &sA[cur * TILE_M * KCHUNK];
#else
    for (int c = 0; c < NCHUNK; ++c) {
        const int kc = c * KCHUNK;
        if (kc + KCHUNK < HDIM) {
            __builtin_prefetch(wrow0 + kc + KCHUNK, 0, 3);
            __builtin_prefetch(wrow1 + kc + KCHUNK, 0, 3);
        }
        __syncthreads();
        // Synchronous staging fallback: 2048 float4s over 256 threads.
        #pragma unroll
        for (int j = 0; j < 8; ++j) {
            int q   = j * 256 + tid;
            int row = q >> 5;
            int c4  = q & 31;
            float4 v = *(const float4*)(xbase + (size_t)row * HDIM + kc + c4 * 4);
            *(float4*)(&sA[row * KCHUNK + c4 * 4]) = v;
        }
        __syncthreads();
        const float* Ab = &sA[0];
#endif

        // K-inner: per 4-wide k step, 2 global B frags reused by 4 A frags
        // -> 8 v_wmma_f32_16x16x4_f32 per 6 loads.
        // A frag layout: lane(l16,h) holds A[m][k+2h], A[m][k+2h+1]
        // B frag layout: lane(l16,h) holds W[n][k+2h], W[n][k+2h+1]
        #pragma unroll 2
        for (int kk = 0; kk < KCHUNK; kk += 4) {
            v2f b0 = *(const v2f*)(wrow0 + kc + kk + 2 * half);
            v2f b1 = *(const v2f*)(wrow1 + kc + kk + 2 * half);
            #pragma unroll
            for (int mt = 0; mt < MSUB; ++mt) {
                v2f a = *(const v2f*)(&Ab[(mt * 16 + l16) * KCHUNK + kk + 2 * half]);
                acc[mt][0] = __builtin_amdgcn_wmma_f32_16x16x4_f32(
                    false, a, false, b0, (short)0, acc[mt][0], false, false);
                acc[mt][1] = __builtin_amdgcn_wmma_f32_16x16x4_f32(
                    false, a, false, b1, (short)0, acc[mt][1], false, false);
            }
        }
#if USE_ASYNC_LDS
        __syncthreads();     // all waves done with buf 'cur' before c+2 overwrites it
#endif
    }

    // C/D layout: VGPR i of lane(l16,h) = D[M = i + 8h][N = l16].
    // Sigmoid and scatter to sScores[token][expert].
    __syncthreads();
    #pragma unroll
    for (int mt = 0; mt < MSUB; ++mt) {
        #pragma unroll
        for (int i = 0; i < 8; ++i) {
            int m = mt * 16 + i + 8 * half;
            sScores[m * NEXP + nb + l16]      = sigmoidf_(acc[mt][0][i]);
            sScores[m * NEXP + nb + 16 + l16] = sigmoidf_(acc[mt][1][i]);
        }
    }
    __syncthreads();

    // --- Gating: one thread per token (threads 0..63 = 2 full waves). ---
    if (tid < TILE_M) {
        const float* row = &sScores[tid * NEXP];

        // Per-group score = sum of top-2 biased scores.
        float gs[NGROUP];
        #pragma unroll
        for (int g = 0; g < NGROUP; ++g) {
            float m1 = -3.0e38f, m2 = -3.0e38f;
            for (int e = 0; e < GSIZE; ++e) {
                float v = row[g * GSIZE + e] + sBias[g * GSIZE + e];
                if (v > m1)      { m2 = m1; m1 = v; }
                else if (v > m2) { m2 = v; }
            }
            gs[g] = m1 + m2;
        }

        // Top-4 groups (ties -> lowest index, matching jax.lax.top_k).
        unsigned gmask = 0;
        for (int it = 0; it < TOPKG; ++it) {
            int best = 0; float bv = -3.0e38f;
            for (int g = 0; g < NGROUP; ++g) {
                if (((gmask >> g) & 1u) == 0 && gs[g] > bv) { bv = gs[g]; best = g; }
            }
            gmask |= 1u << best;
        }

        // Top-8 experts among masked groups by biased score, descending.
        // Key ordering (v desc, index asc); each pass scans keys strictly
        // below the previous pick, so no chosen-bitmask is needed.
        int   idxs[TOPK];
        float wv[TOPK];
        float wsum = 0.0f;
        float pv = 3.0e38f; int pi = -1;
        for (int it = 0; it < TOPK; ++it) {
            float bv = -3.0e38f; int bi = 0;
            for (int g = 0; g < NGROUP; ++g) {
                if (((gmask >> g) & 1u) == 0) continue;
                for (int e = 0; e < GSIZE; ++e) {
                    int ei = g * GSIZE + e;
                    float v = row[ei] + sBias[ei];
                    bool below_prev = (v < pv) || (v == pv && ei > pi);
                    if (!below_prev) continue;
                    bool above_best = (v > bv) || (v == bv && ei < bi);
                    if (above_best) { bv = v; bi = ei; }
                }
            }
            pv = bv; pi = bi;
            idxs[it] = bi;
            float s = row[bi];          // UNbiased sigmoid score
            wv[it]  = s;
            wsum   += s;
        }

        const float scale = 2.5f / (wsum + 1e-20f);
        const int t = tokBase + tid;
        #pragma unroll
        for (int j = 0; j < TOPK; ++j) {
            out_idx[t * TOPK + j] = (float)idxs[j];
            out_w  [t * TOPK + j] = wv[j] * scale;
        }
    }
}

extern "C" void kernel_launch(void* const* d_in, const int* in_sizes, int n_in,
                              void* d_out, int out_size, void* d_ws, size_t ws_size,
                              hipStream_t stream) {
    const float* x    = (const float*)d_in[0];  // hidden_states [4,4096,2048]
    const float* w    = (const float*)d_in[1];  // weight [256,2048]
    const float* bias = (const float*)d_in[2];  // e_score_correction_bias [256]

    float* out      = (float*)d_out;
    float* out_idx  = out;                       // first 16384*8 elements
    float* out_wgt  = out + (size_t)T_TOKENS * TOPK;

    dim3 grid(T_TOKENS / TILE_M);   // 256 blocks of 64 tokens
    dim3 block(256);                // 8 wave32s
    kimi_moe_gate_kernel<<<grid, block, 0, stream>>>(x, w, bias, out_idx, out_wgt);
}